// MSE_15453292331629
// MI455X (gfx1250) — compile-verified
//
#include <hip/hip_runtime.h>

// ---------------------------------------------------------------------------
// Bucketed MSE loss (5 rain-rate buckets) for MI455X / gfx1250.
//
// HBM-bound streaming reduction: 604 MB read once -> ~26 us at 23.3 TB/s.
// Phase 1: grid-stride x2-unrolled float4 NT loads; sse accum in VALU
//          (6 shared v_cmp + cndmask/add per bucket), bucket COUNTS via
//          ballot+popc on the SALU pipe (wave-uniform, co-issues with VALU).
// Phase 2: wave32 shfl-xor tree reduce of the 5 sse partials; lane 0 writes
//          sse + wave counts to LDS table P[16][8] (rows 10..15 zero).
// Phase 3: wave 0 sums the 8 wave-partials with two chained
//          V_WMMA_F32_16X16X4_F32 (B = ones -> row sums over K), then
//          10 global f32 atomics per block into d_ws.
// Phase 4: 1-thread finalize computes sum_b w_b * sse_b / max(cnt_b, 1).
// ---------------------------------------------------------------------------

typedef __attribute__((ext_vector_type(2))) float v2f;
typedef __attribute__((ext_vector_type(4))) float v4f;
typedef __attribute__((ext_vector_type(8))) float v8f;

__global__ void MSE_zero_ws(float* __restrict__ ws) {
    if (threadIdx.x < 10) ws[threadIdx.x] = 0.0f;
}

__device__ __forceinline__ void MSE_accum(float o, float t,
                                          float (&sse)[5], unsigned (&cw)[5]) {
    float d  = o - t;
    float sq = d * d;
    // 6 shared compares; mask combines land on the scalar pipe (lane masks)
    bool ge0   = t >= 0.0f;
    bool ge2   = t >= 2.0f;
    bool ge5   = t >= 5.0f;
    bool ge10  = t >= 10.0f;
    bool ge30  = t >= 30.0f;
    bool lt500 = t < 500.0f;
    bool m[5];
    m[0] = ge0  && !ge2;
    m[1] = ge2  && !ge5;
    m[2] = ge5  && !ge10;
    m[3] = ge10 && !ge30;
    m[4] = ge30 && lt500;
#pragma unroll
    for (int b = 0; b < 5; ++b) {
        sse[b] += m[b] ? sq : 0.0f;                     // VALU: cndmask + add
        cw[b]  += (unsigned)__popcll(__ballot(m[b]));   // SALU: s_bcnt + s_add
    }
}

__global__ __launch_bounds__(256)
void MSE_bucket_kernel(const float* __restrict__ outs,
                       const float* __restrict__ tgts,
                       long long n, float* __restrict__ ws) {
    float    sse[5] = {0.f, 0.f, 0.f, 0.f, 0.f};
    unsigned cw[5]  = {0u, 0u, 0u, 0u, 0u};   // wave-level counts (read on lane 0)

    const long long nv     = n >> 2;
    const long long tid    = (long long)blockIdx.x * blockDim.x + threadIdx.x;
    const long long stride = (long long)gridDim.x * blockDim.x;

    const v4f* __restrict__ o4 = (const v4f*)outs;
    const v4f* __restrict__ t4 = (const v4f*)tgts;

    long long i = tid;
    // 2x unrolled: four 128-bit NT loads in flight per iteration
    for (; i + stride < nv; i += 2 * stride) {
        __builtin_prefetch((const void*)(o4 + i + 2 * stride), 0, 1);
        __builtin_prefetch((const void*)(t4 + i + 2 * stride), 0, 1);
        v4f o0 = __builtin_nontemporal_load(o4 + i);
        v4f t0 = __builtin_nontemporal_load(t4 + i);
        v4f o1 = __builtin_nontemporal_load(o4 + i + stride);
        v4f t1 = __builtin_nontemporal_load(t4 + i + stride);
        MSE_accum(o0.x, t0.x, sse, cw);
        MSE_accum(o0.y, t0.y, sse, cw);
        MSE_accum(o0.z, t0.z, sse, cw);
        MSE_accum(o0.w, t0.w, sse, cw);
        MSE_accum(o1.x, t1.x, sse, cw);
        MSE_accum(o1.y, t1.y, sse, cw);
        MSE_accum(o1.z, t1.z, sse, cw);
        MSE_accum(o1.w, t1.w, sse, cw);
    }
    for (; i < nv; i += stride) {
        v4f ov = __builtin_nontemporal_load(o4 + i);
        v4f tv = __builtin_nontemporal_load(t4 + i);
        MSE_accum(ov.x, tv.x, sse, cw);
        MSE_accum(ov.y, tv.y, sse, cw);
        MSE_accum(ov.z, tv.z, sse, cw);
        MSE_accum(ov.w, tv.w, sse, cw);
    }
    // scalar tail (n % 4 elements)
    for (long long j = (nv << 2) + tid; j < n; j += stride)
        MSE_accum(outs[j], tgts[j], sse, cw);

    // ---- wave32 tree reduction of the 5 sse accumulators ----
#pragma unroll
    for (int b = 0; b < 5; ++b) {
#pragma unroll
        for (int off = 16; off > 0; off >>= 1)
            sse[b] += __shfl_xor(sse[b], off, 32);
    }

    // ---- cross-wave reduction via V_WMMA_F32_16X16X4_F32 ----
    // P[row][wave]: rows 0..4 = sse, rows 5..9 = cnt, rows 10..15 = zero pad.
    __shared__ float P[16][8];
    const int lane = threadIdx.x & 31;
    const int wave = threadIdx.x >> 5;

    if (threadIdx.x < 128) ((float*)P)[threadIdx.x] = 0.0f;
    __syncthreads();
    if (lane == 0) {
#pragma unroll
        for (int b = 0; b < 5; ++b) {
            P[b][wave]     = sse[b];
            P[5 + b][wave] = (float)cw[b];
        }
    }
    __syncthreads();

    if (threadIdx.x < 32) {  // wave-uniform branch: EXEC all-1s for wave 0
        const int row = lane & 15;
        const int hi  = lane >> 4;  // lanes 16..31 supply K=2,3

        v2f ones; ones.x = 1.0f; ones.y = 1.0f;  // B = all ones (4x16)
        v2f a;
        v8f c = {};

        // WMMA #1: sum wave partials 0..3 (A[m][k] = P[m][k])
        a.x = P[row][hi ? 2 : 0];
        a.y = P[row][hi ? 3 : 1];
        c = __builtin_amdgcn_wmma_f32_16x16x4_f32(
                false, a, false, ones, (short)0, c, false, false);

        // WMMA #2: accumulate wave partials 4..7 through C
        a.x = P[row][hi ? 6 : 4];
        a.y = P[row][hi ? 7 : 5];
        c = __builtin_amdgcn_wmma_f32_16x16x4_f32(
                false, a, false, ones, (short)0, c, false, false);

        // D layout: lane 0 comps 0..7 = rows 0..7; lane 16 comps 0..1 = rows 8..9
        if (lane == 0) {
#pragma unroll
            for (int j = 0; j < 8; ++j) atomicAdd(&ws[j], c[j]);
        } else if (lane == 16) {
            atomicAdd(&ws[8], c[0]);
            atomicAdd(&ws[9], c[1]);
        }
    }
}

__global__ void MSE_finalize(const float* __restrict__ ws,
                             float* __restrict__ out) {
    if (threadIdx.x == 0 && blockIdx.x == 0) {
        const float W[5] = {1.0f, 2.0f, 5.0f, 10.0f, 30.0f};
        float s = 0.0f;
#pragma unroll
        for (int b = 0; b < 5; ++b) {
            float sseb = ws[b];
            float cntb = ws[5 + b];
            if (cntb > 0.0f) s += W[b] * sseb / fmaxf(cntb, 1.0f);
        }
        out[0] = s;
    }
}

extern "C" void kernel_launch(void* const* d_in, const int* in_sizes, int n_in,
                              void* d_out, int out_size, void* d_ws, size_t ws_size,
                              hipStream_t stream) {
    const float* outs = (const float*)d_in[0];
    const float* tgts = (const float*)d_in[1];
    float* ws  = (float*)d_ws;
    float* out = (float*)d_out;

    const long long n  = (long long)in_sizes[0];
    const long long nv = n >> 2;

    const int threads = 256;
    long long need = (nv + threads - 1) / threads;
    if (need < 1) need = 1;
    const int blocks = (int)(need < 4096 ? need : 4096);

    MSE_zero_ws<<<1, 32, 0, stream>>>(ws);
    MSE_bucket_kernel<<<blocks, threads, 0, stream>>>(outs, tgts, n, ws);
    MSE_finalize<<<1, 32, 0, stream>>>(ws, out);
}